// VectorQuantizerEMA_46514495816280
// MI455X (gfx1250) — compile-verified
//
#include <hip/hip_runtime.h>

#define DECAYF        0.99f
#define ONE_M_DECAYF  0.01f
#define LOSS_SCALEF   10.0f
#define COMMIT_COEFF  0.25f
#define EPSF          1e-5f

#define N_TOK   16384   // B*S
#define DIM     512
#define KCODES  1024
#define ROWPITCH 516    // 512 + 4 pad: LDS stride 516 dwords -> 16 lanes hit 16 distinct banks

typedef float v2f __attribute__((ext_vector_type(2)));
typedef float v8f __attribute__((ext_vector_type(8)));

// ---- workspace layout (float offsets) ----
#define WS_CNORM   0L                     // [1024]
#define WS_NCNT    1024L                  // [1024]
#define WS_DW      2048L                  // [1024*512]
#define WS_LOSSNUM (2048L + 524288L)      // 526336
#define WS_LOSSDEN 526337L
#define WS_TOTAL   526338L
#define WS_CRAW    526340L                // [1024]
#define WS_IDX     527364L                // [16384] ints
#define WS_ZERO_CNT (WS_LOSSNUM + 4L - WS_NCNT)   // ncnt + dw + lossnum/den/total

__global__ void vq_zero_kernel(float* __restrict__ p, long n) {
    long g = (long)blockIdx.x * blockDim.x + threadIdx.x;
    if (g < n) p[g] = 0.0f;
}

__global__ __launch_bounds__(256) void vq_cnorm_kernel(const float* __restrict__ cb,
                                                       float* __restrict__ cnorm) {
    int k = blockIdx.x;
    const float* row = cb + (long)k * DIM;
    float s = 0.0f;
    for (int d = threadIdx.x; d < DIM; d += 256) { float v = row[d]; s += v * v; }
    __shared__ float red[256];
    red[threadIdx.x] = s; __syncthreads();
    for (int off = 128; off > 0; off >>= 1) {
        if (threadIdx.x < off) red[threadIdx.x] += red[threadIdx.x + off];
        __syncthreads();
    }
    if (threadIdx.x == 0) cnorm[k] = red[0];
}

// One block = 16 token rows. 8 waves; wave w owns codebook column-tiles w, w+8, ...
// processed 4 tiles at a time: one LDS A-fragment load feeds 4 independent WMMA
// accumulation chains (fills the matrix pipe, quarters LDS traffic).
// Gram tile via V_WMMA_F32_16X16X4_F32; argmin proxy = cnorm[k] - 2*G[n,k].
__global__ __launch_bounds__(256) void vq_argmin_kernel(const float* __restrict__ z,
                                                        const float* __restrict__ cb,
                                                        const float* __restrict__ cnorm,
                                                        int* __restrict__ idx_ws,
                                                        float* __restrict__ out_idx) {
    __shared__ float zlds[16 * ROWPITCH];          // 33 KB
    __shared__ float redv[8 * 32 * 8];             // 8 KB
    __shared__ int   redi[8 * 32 * 8];             // 8 KB

    const int tid  = threadIdx.x;
    const int wave = tid >> 5;
    const int lane = tid & 31;
    const int hi   = lane >> 4;        // which K-half of the WMMA operand
    const int ln   = lane & 15;        // M for A, N for B/C/D
    const long rowbase = (long)blockIdx.x * 16;

    // stage 16x512 z strip into LDS (coalesced)
    const float* zt = z + rowbase * DIM;
    for (int i = tid; i < 16 * DIM; i += 256) {
        int r = i >> 9, c = i & 511;
        zlds[r * ROWPITCH + c] = zt[i];
    }
    __syncthreads();

    float minv[8]; int mini[8];
#pragma unroll
    for (int v = 0; v < 8; ++v) { minv[v] = 3.402823466e38f; mini[v] = 0x7fffffff; }

    const float* arow = zlds + ln * ROWPITCH + 2 * hi;     // A: lane = token row

    for (int g = 0; g < 2; ++g) {                          // 2 groups of 4 tiles
        const float* crow[4];
        int colidx[4];
        v8f acc[4];
#pragma unroll
        for (int c = 0; c < 4; ++c) {
            int ct = wave + 8 * (4 * g + c);               // this wave's tile ids
            colidx[c] = ct * 16 + ln;                      // B: lane = codebook row
            crow[c]   = cb + (long)colidx[c] * DIM + 2 * hi;
            acc[c]    = (v8f){};
        }
#pragma unroll 8
        for (int d0 = 0; d0 < DIM; d0 += 4) {
            v2f a; a.x = arow[d0]; a.y = arow[d0 + 1];     // K = d0+2*hi .. +1
            if ((d0 & 31) == 0) {                          // compile-time under unroll 8
#pragma unroll
                for (int c = 0; c < 4; ++c)                // global_prefetch_b8, 1KB ahead
                    if (d0 + 256 < DIM) __builtin_prefetch(crow[c] + d0 + 256, 0, 1);
            }
#pragma unroll
            for (int c = 0; c < 4; ++c) {                  // 4 independent acc chains
                v2f b; b.x = crow[c][d0]; b.y = crow[c][d0 + 1];
                acc[c] = __builtin_amdgcn_wmma_f32_16x16x4_f32(
                    false, a, false, b, (short)0, acc[c], false, false);
            }
        }
#pragma unroll
        for (int c = 0; c < 4; ++c) {
            const float cn = cnorm[colidx[c]];
            const int   ci = colidx[c];
#pragma unroll
            for (int v = 0; v < 8; ++v) {                  // element v: m=v+8*hi, n=ln
                float dval = cn - 2.0f * acc[c][v];
                if (dval < minv[v] || (dval == minv[v] && ci < mini[v])) {
                    minv[v] = dval; mini[v] = ci;
                }
            }
        }
    }
#pragma unroll
    for (int v = 0; v < 8; ++v) {
        redv[(wave * 32 + lane) * 8 + v] = minv[v];
        redi[(wave * 32 + lane) * 8 + v] = mini[v];
    }
    __syncthreads();

    if (tid < 16) {                       // thread t finishes row m = t
        const int laneoff = (tid >= 8) ? 16 : 0;
        const int v = tid & 7;
        float best = 3.402823466e38f; int bidx = 0x7fffffff;
        for (int w = 0; w < 8; ++w)
            for (int l = 0; l < 16; ++l) {
                int s = (w * 32 + laneoff + l) * 8 + v;
                float val = redv[s]; int id = redi[s];
                if (val < best || (val == best && id < bidx)) { best = val; bidx = id; }
            }
        long row = rowbase + tid;
        idx_ws[row]  = bidx;
        out_idx[row] = (float)bidx;       // indices emitted as f32 in concat output
    }
}

// One block per token: gather quantized row -> hidden, scatter-add dw/n, loss partials.
__global__ __launch_bounds__(256) void vq_scatter_kernel(const float* __restrict__ z,
                                                         const float* __restrict__ mask,
                                                         const float* __restrict__ cb,
                                                         const int* __restrict__ idx_ws,
                                                         float* __restrict__ hidden,
                                                         float* __restrict__ dw,
                                                         float* __restrict__ ncnt,
                                                         float* __restrict__ lossnum,
                                                         float* __restrict__ lossden) {
    const long row = blockIdx.x;
    const int  k   = idx_ws[row];
    const float m  = mask[row];
    const float* zr  = z  + row * (long)DIM;
    const float* cr  = cb + (long)k * DIM;
    float*       hr  = hidden + row * (long)DIM;
    float*       dwr = dw + (long)k * DIM;

    float ssq = 0.0f;
    for (int d = threadIdx.x; d < DIM; d += 256) {
        float zv = zr[d], qv = cr[d];
        hr[d] = qv;                       // hidden = z + sg(q - z) == q numerically
        float df = zv - qv;
        ssq += df * df;
        atomicAdd(&dwr[d], zv * m);       // segment-sum of masked inputs
    }
    __shared__ float red[256];
    red[threadIdx.x] = ssq; __syncthreads();
    for (int off = 128; off > 0; off >>= 1) {
        if (threadIdx.x < off) red[threadIdx.x] += red[threadIdx.x + off];
        __syncthreads();
    }
    if (threadIdx.x == 0) {
        atomicAdd(lossnum, (red[0] / (float)DIM) * m);
        atomicAdd(lossden, m);
        atomicAdd(&ncnt[k], m);
    }
}

__global__ __launch_bounds__(256) void vq_count_kernel(const float* __restrict__ ema_count,
                                                       const float* __restrict__ ncnt,
                                                       float* __restrict__ craw,
                                                       float* __restrict__ total) {
    int k = blockIdx.x * 256 + threadIdx.x;
    float v = 0.0f;
    if (k < KCODES) { v = ema_count[k] * DECAYF + ONE_M_DECAYF * ncnt[k]; craw[k] = v; }
    __shared__ float red[256];
    red[threadIdx.x] = v; __syncthreads();
    for (int off = 128; off > 0; off >>= 1) {
        if (threadIdx.x < off) red[threadIdx.x] += red[threadIdx.x + off];
        __syncthreads();
    }
    if (threadIdx.x == 0) atomicAdd(total, red[0]);
}

__global__ __launch_bounds__(256) void vq_finalize_kernel(const float* __restrict__ craw,
                                                          const float* __restrict__ total_p,
                                                          const float* __restrict__ ema_weight,
                                                          const float* __restrict__ dw,
                                                          const float* __restrict__ lossnum,
                                                          const float* __restrict__ lossden,
                                                          float* __restrict__ out_count,
                                                          float* __restrict__ out_weight,
                                                          float* __restrict__ out_loss) {
    long g = (long)blockIdx.x * 256 + threadIdx.x;
    if (g < KCODES) {
        float total = *total_p;
        out_count[g] = (craw[g] + EPSF) / (total + (float)KCODES * EPSF) * total;
    }
    if (g < (long)KCODES * DIM)
        out_weight[g] = ema_weight[g] * DECAYF + ONE_M_DECAYF * dw[g];
    if (g == 0)
        *out_loss = LOSS_SCALEF * COMMIT_COEFF * (*lossnum) / (*lossden);
}

extern "C" void kernel_launch(void* const* d_in, const int* in_sizes, int n_in,
                              void* d_out, int out_size, void* d_ws, size_t ws_size,
                              hipStream_t stream) {
    const float* z          = (const float*)d_in[0];  // [8,2048,512]
    const float* mask       = (const float*)d_in[1];  // [8,2048]
    const float* cb         = (const float*)d_in[2];  // [1024,512]
    const float* ema_count  = (const float*)d_in[3];  // [1024]
    const float* ema_weight = (const float*)d_in[4];  // [1024,512]

    float* out = (float*)d_out;
    float* ws  = (float*)d_ws;

    float* cnorm   = ws + WS_CNORM;
    float* ncnt    = ws + WS_NCNT;
    float* dw      = ws + WS_DW;
    float* lossnum = ws + WS_LOSSNUM;
    float* lossden = ws + WS_LOSSDEN;
    float* total   = ws + WS_TOTAL;
    float* craw    = ws + WS_CRAW;
    int*   idx_ws  = (int*)(ws + WS_IDX);

    // output concat: hidden[N,D], indices[N], loss[1], new_count[K], new_weight[K,D]
    float* out_hidden = out;
    float* out_idx    = out_hidden + (long)N_TOK * DIM;
    float* out_loss   = out_idx + N_TOK;
    float* out_count  = out_loss + 1;
    float* out_weight = out_count + KCODES;

    vq_zero_kernel<<<(int)((WS_ZERO_CNT + 255) / 256), 256, 0, stream>>>(ncnt, WS_ZERO_CNT);
    vq_cnorm_kernel<<<KCODES, 256, 0, stream>>>(cb, cnorm);
    vq_argmin_kernel<<<N_TOK / 16, 256, 0, stream>>>(z, cb, cnorm, idx_ws, out_idx);
    vq_scatter_kernel<<<N_TOK, 256, 0, stream>>>(z, mask, cb, idx_ws, out_hidden,
                                                 dw, ncnt, lossnum, lossden);
    vq_count_kernel<<<(KCODES + 255) / 256, 256, 0, stream>>>(ema_count, ncnt, craw, total);
    vq_finalize_kernel<<<(int)(((long)KCODES * DIM + 255) / 256), 256, 0, stream>>>(
        craw, total, ema_weight, dw, lossnum, lossden, out_count, out_weight, out_loss);
}